// OneByOneMerge5_2D_20538533609780
// MI455X (gfx1250) — compile-verified
//
#include <hip/hip_runtime.h>

// ---------------------------------------------------------------------------
// OneByOneMerge5: out[b,h,w,c] = sum_{j=0..4} w[j,c] * t_j[b,h,w,c]
// Shapes: t_j = (8,128,128,128) fp32, w = (5,128) fp32.
//
// Memory-bound streaming kernel (AI ~ 0.37 FLOP/B; HBM floor ~17.3 us at
// 23.3 TB/s). Strategy:
//   * float4 (128-bit) global traffic -> global_load_b128 / global_store_b128
//   * non-temporal hints (CDNA5 TH bits): 402 MB streamed once > 192 MB L2,
//     so bypass cache retention (th:TH_LOAD_NT / th:TH_STORE_NT)
//   * wave32 trick: C=128 = 32 lanes x 4 floats -> each lane owns a fixed
//     channel quad; the 5 weight float4s live in registers for the whole
//     kernel (no LDS, no reloads)
//   * x4 unroll -> up to 20 outstanding 16B loads per thread for MLP
// WMMA is deliberately NOT used: the weight matrix is diagonal per channel;
// a 16x16 WMMA tile would waste 15/16 of its MACs and move the same bytes.
// ---------------------------------------------------------------------------

typedef float v4f __attribute__((ext_vector_type(4)));

#define BLOCK 256
#define UNROLL 4

static constexpr unsigned kC = 128;          // channels
static constexpr unsigned kQuadsPerRow = 32; // 128 / 4 == wave32 lanes

__global__ __launch_bounds__(BLOCK) void merge5_nt_kernel(
    const v4f* __restrict__ t1, const v4f* __restrict__ t2,
    const v4f* __restrict__ t3, const v4f* __restrict__ t4,
    const v4f* __restrict__ t5, const float* __restrict__ w,
    v4f* __restrict__ out, unsigned nquads)
{
    const unsigned tid    = blockIdx.x * BLOCK + threadIdx.x;
    const unsigned stride = gridDim.x * BLOCK;            // multiple of 32
    const unsigned lane   = threadIdx.x & (kQuadsPerRow - 1u);

    // q = tid + k*stride  ==>  q % 32 == lane for every iteration, so each
    // lane's channel quad c = 4*lane is loop-invariant. Load the 5 weight
    // quads ONCE into registers (16B aligned: lane*4 floats).
    const v4f w1 = *reinterpret_cast<const v4f*>(w + 0u * kC + lane * 4u);
    const v4f w2 = *reinterpret_cast<const v4f*>(w + 1u * kC + lane * 4u);
    const v4f w3 = *reinterpret_cast<const v4f*>(w + 2u * kC + lane * 4u);
    const v4f w4 = *reinterpret_cast<const v4f*>(w + 3u * kC + lane * 4u);
    const v4f w5 = *reinterpret_cast<const v4f*>(w + 4u * kC + lane * 4u);

    unsigned q = tid;

    // Main x4-unrolled stream: issue all 20 NT b128 loads up front (MLP),
    // then 4 independent FMA chains, then 4 NT b128 stores.
    for (; q + 3u * stride < nquads; q += 4u * stride) {
        const unsigned q0 = q, q1 = q + stride, q2 = q + 2u * stride, q3 = q + 3u * stride;

        v4f a0 = __builtin_nontemporal_load(t1 + q0);
        v4f a1 = __builtin_nontemporal_load(t1 + q1);
        v4f a2 = __builtin_nontemporal_load(t1 + q2);
        v4f a3 = __builtin_nontemporal_load(t1 + q3);
        v4f b0 = __builtin_nontemporal_load(t2 + q0);
        v4f b1 = __builtin_nontemporal_load(t2 + q1);
        v4f b2 = __builtin_nontemporal_load(t2 + q2);
        v4f b3 = __builtin_nontemporal_load(t2 + q3);
        v4f c0 = __builtin_nontemporal_load(t3 + q0);
        v4f c1 = __builtin_nontemporal_load(t3 + q1);
        v4f c2 = __builtin_nontemporal_load(t3 + q2);
        v4f c3 = __builtin_nontemporal_load(t3 + q3);
        v4f d0 = __builtin_nontemporal_load(t4 + q0);
        v4f d1 = __builtin_nontemporal_load(t4 + q1);
        v4f d2 = __builtin_nontemporal_load(t4 + q2);
        v4f d3 = __builtin_nontemporal_load(t4 + q3);
        v4f e0 = __builtin_nontemporal_load(t5 + q0);
        v4f e1 = __builtin_nontemporal_load(t5 + q1);
        v4f e2 = __builtin_nontemporal_load(t5 + q2);
        v4f e3 = __builtin_nontemporal_load(t5 + q3);

        v4f r0 = a0 * w1; r0 += b0 * w2; r0 += c0 * w3; r0 += d0 * w4; r0 += e0 * w5;
        v4f r1 = a1 * w1; r1 += b1 * w2; r1 += c1 * w3; r1 += d1 * w4; r1 += e1 * w5;
        v4f r2 = a2 * w1; r2 += b2 * w2; r2 += c2 * w3; r2 += d2 * w4; r2 += e2 * w5;
        v4f r3 = a3 * w1; r3 += b3 * w2; r3 += c3 * w3; r3 += d3 * w4; r3 += e3 * w5;

        __builtin_nontemporal_store(r0, out + q0);
        __builtin_nontemporal_store(r1, out + q1);
        __builtin_nontemporal_store(r2, out + q2);
        __builtin_nontemporal_store(r3, out + q3);
    }

    // Tail (empty for the reference shape: 4,194,304 quads / (4096*256*4) exact).
    for (; q < nquads; q += stride) {
        v4f a = __builtin_nontemporal_load(t1 + q);
        v4f b = __builtin_nontemporal_load(t2 + q);
        v4f c = __builtin_nontemporal_load(t3 + q);
        v4f d = __builtin_nontemporal_load(t4 + q);
        v4f e = __builtin_nontemporal_load(t5 + q);
        v4f r = a * w1; r += b * w2; r += c * w3; r += d * w4; r += e * w5;
        __builtin_nontemporal_store(r, out + q);
    }
}

extern "C" void kernel_launch(void* const* d_in, const int* in_sizes, int n_in,
                              void* d_out, int out_size, void* d_ws, size_t ws_size,
                              hipStream_t stream) {
    (void)in_sizes; (void)n_in; (void)d_ws; (void)ws_size;

    const v4f*  t1 = reinterpret_cast<const v4f*>(d_in[0]);
    const v4f*  t2 = reinterpret_cast<const v4f*>(d_in[1]);
    const v4f*  t3 = reinterpret_cast<const v4f*>(d_in[2]);
    const v4f*  t4 = reinterpret_cast<const v4f*>(d_in[3]);
    const v4f*  t5 = reinterpret_cast<const v4f*>(d_in[4]);
    const float* w = reinterpret_cast<const float*>(d_in[5]);
    v4f*       out = reinterpret_cast<v4f*>(d_out);

    const unsigned nquads = static_cast<unsigned>(out_size) / 4u; // 4,194,304
    const unsigned quads_per_block = BLOCK * UNROLL;              // 1024
    const unsigned blocks = (nquads + quads_per_block - 1u) / quads_per_block; // 4096

    merge5_nt_kernel<<<blocks, BLOCK, 0, stream>>>(t1, t2, t3, t4, t5, w, out, nquads);
}